// GATConvolution_85401129714252
// MI455X (gfx1250) — compile-verified
//
#include <hip/hip_runtime.h>
#include <hip/hip_bf16.h>
#include <math.h>

#define N_NODES 100000
#define N_EDGES 3200000
#define NEG_SLOPE 0.2f

typedef __attribute__((ext_vector_type(2))) float v2f;
typedef __attribute__((ext_vector_type(8))) float v8f;

__device__ __forceinline__ float leakyf(float v) { return v > 0.f ? v : NEG_SLOPE * v; }

// Monotone float atomic-max via int atomics (init value must be -inf).
__device__ __forceinline__ void atomicMaxF(float* addr, float v) {
    if (v >= 0.f) atomicMax((int*)addr, __float_as_int(v));
    else          atomicMin((unsigned int*)addr, __float_as_uint(v));
}

// ---------------------------------------------------------------------------
// Templated fp32 WMMA GEMM:  C[N x MOUT] = act(A[N x K] @ W[K x M] + bias)
// M is a multiple of 16 (weights pre-padded); MOUT <= M is the stored width.
// One wave per 16-row tile; V_WMMA_F32_16X16X4_F32; fully unrolled k-chain;
// A fragments loaded once and reused across all M/16 column tiles.
// ACT: 0 = identity, 1 = relu
// ---------------------------------------------------------------------------
template <int K, int M, int MOUT, int ACT>
__global__ __launch_bounds__(256)
void gemm_wmma_f32(const float* __restrict__ A, const float* __restrict__ W,
                   const float* __restrict__ bias, float* __restrict__ C,
                   int Nrows)
{
    const int lane     = threadIdx.x & 31;
    const int wave     = threadIdx.x >> 5;
    const int row_base = (blockIdx.x * 8 + wave) * 16;
    if (row_base >= Nrows) return;           // wave-uniform: EXEC stays all-ones

    const int nlo   = lane & 15;             // A row within tile / B,C column within tile
    const int khalf = (lane >> 4) * 2;       // K sub-offset selected by lane half
    const int arow  = row_base + nlo;
    const int rofs  = (lane >> 4) * 8;       // D row offset for this lane half

    // Load all A fragments for this row tile (reused by every column tile).
    v2f av[K / 4];
    #pragma unroll
    for (int kc = 0; kc < K / 4; ++kc) {
        const int k0 = kc * 4 + khalf;
        av[kc].x = A[arow * K + k0];
        av[kc].y = A[arow * K + k0 + 1];
    }

    #pragma unroll
    for (int mt = 0; mt < M / 16; ++mt) {
        const int col = mt * 16 + nlo;
        const float bv = bias ? bias[col] : 0.0f;
        v8f c;
        #pragma unroll
        for (int v = 0; v < 8; ++v) c[v] = bv;

        #pragma unroll
        for (int kc = 0; kc < K / 4; ++kc) {
            const int k0 = kc * 4 + khalf;
            v2f b;
            b.x = W[k0 * M + col];
            b.y = W[(k0 + 1) * M + col];
            // (neg_a, A, neg_b, B, c_mod, C, reuse_a, reuse_b)
            c = __builtin_amdgcn_wmma_f32_16x16x4_f32(
                    false, av[kc], false, b, (short)0, c, false, false);
        }

        if (MOUT == M || col < MOUT) {
            #pragma unroll
            for (int v = 0; v < 8; ++v) {
                float r = c[v];
                if (ACT == 1) r = r > 0.f ? r : 0.f;
                C[(row_base + rofs + v) * MOUT + col] = r;
            }
        }
    }
}

// ---------------------------------------------------------------------------
// Zero-pad final linear layer: W3 [64,4] -> [64,16], b3 [4] -> [16].
// ---------------------------------------------------------------------------
__global__ void pad_w3_kernel(const float* __restrict__ W3, const float* __restrict__ b3,
                              float* __restrict__ W3p, float* __restrict__ b3p)
{
    int t = blockIdx.x * blockDim.x + threadIdx.x;
    if (t < 64 * 16) {
        int k = t >> 4, c = t & 15;
        W3p[t] = (c < 4) ? W3[k * 4 + c] : 0.0f;
    }
    if (t < 16) b3p[t] = (t < 4) ? b3[t] : 0.0f;
}

// ---------------------------------------------------------------------------
// wae[i][j] = sum_k We[i][j][k] * ae[i][k]   (3 layers x 8)
// ---------------------------------------------------------------------------
__global__ void wae_kernel(const float* __restrict__ We, const float* __restrict__ ae,
                           float* __restrict__ wae)
{
    int t = threadIdx.x;
    if (t < 24) {
        int i = t >> 3, j = t & 7;
        float s = 0.f;
        #pragma unroll
        for (int k = 0; k < 16; ++k) s += We[i * 128 + j * 16 + k] * ae[i * 16 + k];
        wae[t] = s;
    }
}

// ---------------------------------------------------------------------------
// Per-node prep: attention scores from h, init segment-max/denom, out = bias.
// ---------------------------------------------------------------------------
__global__ __launch_bounds__(256)
void prep_nodes_kernel(const float* __restrict__ h,
                       const float* __restrict__ asrc, const float* __restrict__ adst,
                       const float* __restrict__ b,
                       float* __restrict__ ssrc, float* __restrict__ sdst,
                       float* __restrict__ m, float* __restrict__ den,
                       float* __restrict__ outb)
{
    int n = blockIdx.x * blockDim.x + threadIdx.x;
    if (n >= N_NODES) return;
    float ss = 0.f, sd = 0.f;
    #pragma unroll
    for (int f = 0; f < 16; ++f) {
        float hv = h[n * 16 + f];
        ss += hv * asrc[f];
        sd += hv * adst[f];
        outb[n * 16 + f] = b[f];
    }
    ssrc[n] = ss;
    sdst[n] = sd;
    m[n]    = -INFINITY;
    den[n]  = 0.f;
}

// ---------------------------------------------------------------------------
// Pass 1 over edges: logit = leaky(s_src[src] + s_dst[dst] + ea.wae), seg-max.
// ---------------------------------------------------------------------------
__global__ __launch_bounds__(256)
void edge_logit_max_kernel(const int* __restrict__ src, const int* __restrict__ dst,
                           const float* __restrict__ ea, const float* __restrict__ wae,
                           const float* __restrict__ ssrc, const float* __restrict__ sdst,
                           float* __restrict__ lp, float* __restrict__ m)
{
    int e = blockIdx.x * blockDim.x + threadIdx.x;
    if (e >= N_EDGES) return;
    int s = src[e], d = dst[e];
    float se = 0.f;
    #pragma unroll
    for (int j = 0; j < 8; ++j) se += ea[e * 8 + j] * wae[j];
    float lg = leakyf(ssrc[s] + sdst[d] + se);
    lp[e] = lg;
    atomicMaxF(&m[d], lg);
}

// ---------------------------------------------------------------------------
// Pass 2 over edges: p = exp(logit - m[dst]), seg-sum into denom.
// ---------------------------------------------------------------------------
__global__ __launch_bounds__(256)
void edge_exp_sum_kernel(const int* __restrict__ dst, float* __restrict__ lp,
                         const float* __restrict__ m, float* __restrict__ den)
{
    int e = blockIdx.x * blockDim.x + threadIdx.x;
    if (e >= N_EDGES) return;
    int d = dst[e];
    float p = __expf(lp[e] - m[d]);
    lp[e] = p;
    atomicAdd(&den[d], p);
}

// ---------------------------------------------------------------------------
// Pass 3: out[dst] += h[src] * (p / (denom[dst]+eps)); one thread per (edge,f).
// ---------------------------------------------------------------------------
__global__ __launch_bounds__(256)
void edge_aggregate_kernel(const int* __restrict__ src, const int* __restrict__ dst,
                           const float* __restrict__ lp, const float* __restrict__ den,
                           const float* __restrict__ h, float* __restrict__ outb)
{
    long long t = (long long)blockIdx.x * blockDim.x + threadIdx.x;
    if (t >= (long long)N_EDGES * 16) return;
    int e = (int)(t >> 4);
    int f = (int)(t & 15);
    int d = dst[e], s = src[e];
    float alpha = lp[e] / (den[d] + 1e-16f);
    atomicAdd(&outb[d * 16 + f], h[s * 16 + f] * alpha);
}

// ---------------------------------------------------------------------------
// In-place leaky ReLU over node features.
// ---------------------------------------------------------------------------
__global__ __launch_bounds__(256)
void post_leaky_kernel(float* __restrict__ v, int n)
{
    int i = blockIdx.x * blockDim.x + threadIdx.x;
    if (i < n) v[i] = leakyf(v[i]);
}

// ---------------------------------------------------------------------------
extern "C" void kernel_launch(void* const* d_in, const int* in_sizes, int n_in,
                              void* d_out, int out_size, void* d_ws, size_t ws_size,
                              hipStream_t stream)
{
    const float* x     = (const float*)d_in[0];
    const int*   ei    = (const int*)  d_in[1];   // [2, E]: row0 = src, row1 = dst
    const float* ea    = (const float*)d_in[2];
    const float* cW    = (const float*)d_in[3];   // [3,16,16]
    const float* casrc = (const float*)d_in[4];   // [3,16]
    const float* cadst = (const float*)d_in[5];   // [3,16]
    const float* cWe   = (const float*)d_in[6];   // [3,8,16]
    const float* cae   = (const float*)d_in[7];   // [3,16]
    const float* cb    = (const float*)d_in[8];   // [3,16]
    const float* W0 = (const float*)d_in[9];  const float* b0 = (const float*)d_in[10];
    const float* W1 = (const float*)d_in[11]; const float* b1 = (const float*)d_in[12];
    const float* W2 = (const float*)d_in[13]; const float* b2 = (const float*)d_in[14];
    const float* W3 = (const float*)d_in[15]; const float* b3 = (const float*)d_in[16];
    float* out = (float*)d_out;

    const int* e_src = ei;
    const int* e_dst = ei + N_EDGES;

    // Workspace layout (floats)
    float* ws   = (float*)d_ws;
    float* bufA = ws;                          // N*16  : h = x @ W
    float* bufB = bufA + (size_t)N_NODES * 16; // N*16  : aggregation out -> next x
    float* ssrc = bufB + (size_t)N_NODES * 16; // N
    float* sdst = ssrc + N_NODES;              // N
    float* segm = sdst + N_NODES;              // N
    float* den  = segm + N_NODES;              // N
    float* lp   = den  + N_NODES;              // E     : logits then p
    float* wae  = lp   + N_EDGES;              // 32
    float* w3p  = wae  + 32;                   // 64*16 : zero-padded lin_W3
    float* b3p  = w3p  + 64 * 16;              // 16    : zero-padded lin_b3
    float* mlpA = b3p  + 16;                   // N*64
    float* mlpB = bufA;                        // alias: conv scratch reused (6.8M floats >= 6.4M)

    const int nodeBlocks  = (N_NODES + 255) / 256;
    const int gemmBlocks  = (N_NODES + 127) / 128;   // 8 waves x 16 rows per block
    const int edgeBlocks  = (N_EDGES + 255) / 256;
    const long long aggT  = (long long)N_EDGES * 16;
    const int aggBlocks   = (int)((aggT + 255) / 256);
    const int featBlocks  = (N_NODES * 16 + 255) / 256;

    wae_kernel<<<1, 32, 0, stream>>>(cWe, cae, wae);
    pad_w3_kernel<<<4, 256, 0, stream>>>(W3, b3, w3p, b3p);

    const float* xin = x;
    for (int i = 0; i < 3; ++i) {
        gemm_wmma_f32<16, 16, 16, 0><<<gemmBlocks, 256, 0, stream>>>(
            xin, cW + i * 256, nullptr, bufA, N_NODES);
        prep_nodes_kernel<<<nodeBlocks, 256, 0, stream>>>(
            bufA, casrc + i * 16, cadst + i * 16, cb + i * 16,
            ssrc, sdst, segm, den, bufB);
        edge_logit_max_kernel<<<edgeBlocks, 256, 0, stream>>>(
            e_src, e_dst, ea, wae + i * 8, ssrc, sdst, lp, segm);
        edge_exp_sum_kernel<<<edgeBlocks, 256, 0, stream>>>(
            e_dst, lp, segm, den);
        edge_aggregate_kernel<<<aggBlocks, 256, 0, stream>>>(
            e_src, e_dst, lp, den, bufA, bufB);
        post_leaky_kernel<<<featBlocks, 256, 0, stream>>>(bufB, N_NODES * 16);
        xin = bufB;
    }

    // MLP head: 16 -> 64 -> 64 -> 64 -> 4 (last layer padded to width 16)
    gemm_wmma_f32<16, 64, 64, 1><<<gemmBlocks, 256, 0, stream>>>(bufB, W0, b0, mlpA, N_NODES);
    gemm_wmma_f32<64, 64, 64, 1><<<gemmBlocks, 256, 0, stream>>>(mlpA, W1, b1, mlpB, N_NODES);
    gemm_wmma_f32<64, 64, 64, 1><<<gemmBlocks, 256, 0, stream>>>(mlpB, W2, b2, mlpA, N_NODES);
    gemm_wmma_f32<64, 16, 4, 0><<<gemmBlocks, 256, 0, stream>>>(mlpA, w3p, b3p, out, N_NODES);
}